// MCPRetriever_25598005084905
// MI455X (gfx1250) — compile-verified
//
#include <hip/hip_runtime.h>
#include <math.h>
#include <stdint.h>

// ---------------- problem constants ----------------
#define DD      512
#define BB      16
#define SS      512
#define NDOCS   500000
#define NTILES  31250            // NDOCS / 16
#define TOPK    8

#define K3_BLOCKS 256
#define K3_WPB    8              // waves per block (256 threads)
#define K3_WAVES  (K3_BLOCKS * K3_WPB)   // 2048

#define QSTRIDE 516              // effective LDS row stride for q_n (conflict-free)
#define SPAD    20               // LDS row stride for sim staging (conflict-free)

typedef __attribute__((ext_vector_type(2))) float v2f;
typedef __attribute__((ext_vector_type(8))) float v8f;
typedef __attribute__((ext_vector_type(4))) unsigned int u32x4;
typedef __attribute__((ext_vector_type(8))) int i32x8;
typedef __attribute__((ext_vector_type(4))) int i32x4;

__device__ __forceinline__ v8f wmma_f32_k4(v2f a, v2f b, v8f c) {
    // D(16x16,f32) = A(16x4,f32) * B(4x16,f32) + C
    return __builtin_amdgcn_wmma_f32_16x16x4_f32(
        /*neg_a=*/false, a, /*neg_b=*/false, b,
        /*c_mod=*/(short)0, c, /*reuse_a=*/false, /*reuse_b=*/false);
}

// ------------------------------------------------------------------
// k1: h = relu(query @ W1^T + b1); per-WG 64-row column partial sums.
//     grid=(4 colgroups, 128 msupers), block=128 (4 waves).
//     Each wave keeps its 16 query rows hot across 8 sequential n-tiles.
// ------------------------------------------------------------------
__global__ void k1_mlp1(const float* __restrict__ query,
                        const float* __restrict__ W1,
                        const float* __restrict__ b1,
                        float* __restrict__ part /*[128][512]*/) {
    const int cg     = blockIdx.x;            // 0..3  (128 columns each)
    const int msuper = blockIdx.y;            // 0..127 (64 rows each)
    const int wave   = threadIdx.x >> 5;      // 0..3
    const int lane   = threadIdx.x & 31;
    const int l16    = lane & 15;
    const int hi     = lane >> 4;             // 0/1
    const int m0     = msuper * 64 + wave * 16;

    const float* arow = query + (size_t)(m0 + l16) * DD + 2 * hi;   // A[m][k]
    __shared__ float red[4 * 16];

    for (int nt2 = 0; nt2 < 8; ++nt2) {
        const int n = cg * 128 + nt2 * 16 + l16;
        const float* brow = W1 + (size_t)n * DD + 2 * hi;           // B[k][n] = W1[n][k]

        v8f acc = {};
#pragma unroll 4
        for (int k0 = 0; k0 < DD; k0 += 4) {
            v2f a = *(const v2f*)(arow + k0);
            v2f b = *(const v2f*)(brow + k0);
            acc = wmma_f32_k4(a, b, acc);
        }

        const float bias = b1[n];
        float psum = 0.f;
#pragma unroll
        for (int r = 0; r < 8; ++r) {
            float h = acc[r] + bias;          // row M = r + 8*hi of this tile
            psum += (h > 0.f) ? h : 0.f;
        }
        psum += __shfl_xor(psum, 16);         // full 16-row sum for column n

        if (hi == 0) red[wave * 16 + l16] = psum;
        __syncthreads();
        if (threadIdx.x < 16) {
            part[msuper * DD + cg * 128 + nt2 * 16 + threadIdx.x] =
                red[threadIdx.x] + red[16 + threadIdx.x] +
                red[32 + threadIdx.x] + red[48 + threadIdx.x];
        }
        __syncthreads();                      // protect red[] before next n-tile
    }
}

// k1b: hbar[b][k] = (sum_{j<8} part[b*8+j][k]) / 512
__global__ void k1b_reduce(const float* __restrict__ part, float* __restrict__ hbar) {
    int i = blockIdx.x * blockDim.x + threadIdx.x;   // 0..8191
    int b = i >> 9, k = i & 511;
    float s = 0.f;
#pragma unroll
    for (int j = 0; j < 8; ++j) s += part[((b << 3) + j) * DD + k];
    hbar[i] = s * (1.0f / (float)SS);
}

// ------------------------------------------------------------------
// k2: q_vec = hbar @ W2^T + b2.  grid=32 (one wave each).
// ------------------------------------------------------------------
__global__ void k2_mlp2(const float* __restrict__ hbar,
                        const float* __restrict__ W2,
                        const float* __restrict__ b2,
                        float* __restrict__ qvec) {
    const int ntile = blockIdx.x;
    const int lane  = threadIdx.x & 31;
    const int l16   = lane & 15;
    const int hi    = lane >> 4;
    const int n     = ntile * 16 + l16;

    const float* arow = hbar + (size_t)l16 * DD + 2 * hi;   // A[m=batch][k]
    const float* brow = W2   + (size_t)n * DD + 2 * hi;     // B[k][n] = W2[n][k]

    v8f acc = {};
#pragma unroll 4
    for (int k0 = 0; k0 < DD; k0 += 4) {
        v2f a = *(const v2f*)(arow + k0);
        v2f b = *(const v2f*)(brow + k0);
        acc = wmma_f32_k4(a, b, acc);
    }
    const float bias = b2[n];
#pragma unroll
    for (int r = 0; r < 8; ++r) {
        int b = r + 8 * hi;                                  // batch
        qvec[b * DD + n] = acc[r] + bias;
    }
}

// k2b: q_n = q_vec / max(||q_vec||, 1e-8), one block per batch
__global__ void k2b_norm(const float* __restrict__ qvec, float* __restrict__ qn) {
    const int b = blockIdx.x, tid = threadIdx.x;
    float v0 = qvec[b * DD + tid];
    float v1 = qvec[b * DD + tid + 256];
    __shared__ float red[256];
    red[tid] = v0 * v0 + v1 * v1;
    __syncthreads();
    for (int n = 128; n; n >>= 1) {
        if (tid < n) red[tid] += red[tid + n];
        __syncthreads();
    }
    float rn = 1.0f / fmaxf(sqrtf(red[0]), 1e-8f);
    qn[b * DD + tid]       = v0 * rn;
    qn[b * DD + tid + 256] = v1 * rn;
}

// ------------------------------------------------------------------
// k3: sims = q_n @ d_n^T via WMMA f32, fused per-wave top-8 per batch.
//     q_n is brought into LDS by the Tensor Data Mover (TDM) with the
//     pad feature producing a conflict-free 516-float row stride:
//       lds_off(b,k) = b*516 + k + 2*(k>>8)
//     grid = 256 blocks x 256 threads (8 waves) = 2048 waves.
// ------------------------------------------------------------------
__global__ void k3_sims(const float* __restrict__ qn,
                        const float* __restrict__ doc,
                        float* __restrict__ cand_s,
                        int*   __restrict__ cand_i) {
    __shared__ float qs[BB * QSTRIDE];         // 33 KB (padded q_n)
    __shared__ float stage[K3_WPB][16 * SPAD]; // 10 KB (per-wave 16 docs x 16 b)

#if __has_builtin(__builtin_amdgcn_tensor_load_to_lds) && __has_builtin(__builtin_amdgcn_s_wait_tensorcnt)
    if (threadIdx.x < 32) {                    // wave 0 issues one TDM descriptor
        typedef __attribute__((address_space(3))) float* lds_fp;
        unsigned lds_off = (unsigned)(uintptr_t)(lds_fp)&qs[0];
        unsigned long long ga = (unsigned long long)(uintptr_t)qn;
        u32x4 g0; i32x8 g1; i32x4 g2; i32x4 g3; i32x8 g4;
        // D# group 0: count=1, lds_addr, global_addr[56:0], type=2
        g0[0] = 1u;
        g0[1] = lds_off;
        g0[2] = (unsigned)(ga & 0xffffffffu);
        g0[3] = (unsigned)((ga >> 32) & 0x01ffffffu) | (2u << 30);
        // D# group 1: data_size=4B, pad_enable, pad every 256 DW by 2 DW
        g1[0] = (2 << 16) | (1 << 20) | (7 << 22) | (1 << 25);
        g1[1] = (int)(512u << 16);             // tensor_dim0 = 512
        g1[2] = (int)(16u  << 16);             // tensor_dim1 = 16
        g1[3] = (int)(512u << 16);             // tile_dim0   = 512
        g1[4] = 16;                            // tile_dim1=16, tile_dim2=0
        g1[5] = 512;                           // tensor_dim0_stride = 512
        g1[6] = 0; g1[7] = 0;
        g2[0] = g2[1] = g2[2] = g2[3] = 0;     // 2-D tensor: groups 2/3 unused
        g3[0] = g3[1] = g3[2] = g3[3] = 0;
        g4[0] = g4[1] = g4[2] = g4[3] = 0;
        g4[4] = g4[5] = g4[6] = g4[7] = 0;
        __builtin_amdgcn_tensor_load_to_lds(g0, g1, g2, g3, g4, 0);
        __builtin_amdgcn_s_wait_tensorcnt(0);
    }
#else
    for (int i = threadIdx.x; i < BB * DD; i += blockDim.x) {
        int b = i >> 9, k = i & 511;
        qs[b * QSTRIDE + k + 2 * (k >> 8)] = qn[i];
    }
#endif
    __syncthreads();

    const int wave = threadIdx.x >> 5;
    const int lane = threadIdx.x & 31;
    const int l16  = lane & 15;
    const int hi   = lane >> 4;
    const int gwave = blockIdx.x * K3_WPB + wave;

    float ts[TOPK]; int ti[TOPK];
#pragma unroll
    for (int j = 0; j < TOPK; ++j) { ts[j] = -3.4e38f; ti[j] = 0; }

    const float* qrowA = qs + l16 * QSTRIDE + 2 * hi;   // k in [0,256)
    const float* qrowB = qrowA + 2;                     // k in [256,512): skip pad
    float* st = &stage[wave][0];

    for (int t = gwave; t < NTILES; t += K3_WAVES) {
        const float* drow = doc + ((size_t)t * 16 + l16) * DD + 2 * hi; // B[k][n=doc]
        v8f acc = {};
        float ss = 0.f;
#pragma unroll 4
        for (int k0 = 0; k0 < 256; k0 += 4) {
            v2f a = *(const v2f*)(qrowA + k0);
            v2f b = *(const v2f*)(drow + k0);
            ss = fmaf(b.x, b.x, ss);
            ss = fmaf(b.y, b.y, ss);
            acc = wmma_f32_k4(a, b, acc);
        }
#pragma unroll 4
        for (int k0 = 256; k0 < 512; k0 += 4) {
            v2f a = *(const v2f*)(qrowB + k0);
            v2f b = *(const v2f*)(drow + k0);
            ss = fmaf(b.x, b.x, ss);
            ss = fmaf(b.y, b.y, ss);
            acc = wmma_f32_k4(a, b, acc);
        }
        ss += __shfl_xor(ss, 16);                       // full ||doc_{l16}||^2
        float rs = 1.0f / fmaxf(sqrtf(ss), 1e-8f);

        // stage[doc][batch] = sim ;  lane holds (batch=r+8*hi, doc=l16)
#pragma unroll
        for (int r = 0; r < 8; ++r)
            st[l16 * SPAD + r + 8 * hi] = acc[r] * rs;

        // lane owns batch=l16, docs j = 8*hi .. 8*hi+7 of this tile
        const int docbase = t * 16;
#pragma unroll
        for (int j8 = 0; j8 < 8; ++j8) {
            int j = j8 + 8 * hi;
            float v = st[j * SPAD + l16];
            int   id = docbase + j;
            if (v > ts[TOPK - 1]) {
#pragma unroll
                for (int p = 0; p < TOPK; ++p) {
                    bool gt = v > ts[p];
                    float nf = gt ? ts[p] : v;     int ni = gt ? ti[p] : id;
                    ts[p]    = gt ? v     : ts[p]; ti[p] = gt ? id   : ti[p];
                    v = nf; id = ni;
                }
            }
        }
    }

    const int base = (gwave * 32 + lane) * TOPK;
#pragma unroll
    for (int j = 0; j < TOPK; ++j) { cand_s[base + j] = ts[j]; cand_i[base + j] = ti[j]; }
}

// ------------------------------------------------------------------
// k4: per-batch merge of candidates -> sorted top-8, then fusion:
//     softmax weights, gather, sigmoid gate, evidence.  16 blocks x 256.
// ------------------------------------------------------------------
__global__ void k4_merge_fuse(const float* __restrict__ cand_s,
                              const int*   __restrict__ cand_i,
                              const float* __restrict__ doc,
                              const float* __restrict__ Wg,
                              const float* __restrict__ bg,
                              float* __restrict__ out) {
    const int b = blockIdx.x, tid = threadIdx.x;
    __shared__ float ls[256 * TOPK];
    __shared__ int   li[256 * TOPK];

    float ts[TOPK]; int ti[TOPK];
#pragma unroll
    for (int j = 0; j < TOPK; ++j) { ts[j] = -3.4e38f; ti[j] = 0; }

    for (int w = tid; w < K3_WAVES; w += 256) {
#pragma unroll
        for (int half = 0; half < 2; ++half) {
            const int base = (w * 32 + b + 16 * half) * TOPK;
#pragma unroll
            for (int j = 0; j < TOPK; ++j) {
                float v = cand_s[base + j];
                int   id = cand_i[base + j];
                if (v > ts[TOPK - 1]) {
#pragma unroll
                    for (int p = 0; p < TOPK; ++p) {
                        bool gt = v > ts[p];
                        float nf = gt ? ts[p] : v;   int ni = gt ? ti[p] : id;
                        ts[p]    = gt ? v : ts[p];   ti[p] = gt ? id : ti[p];
                        v = nf; id = ni;
                    }
                }
            }
        }
    }
#pragma unroll
    for (int j = 0; j < TOPK; ++j) { ls[tid * TOPK + j] = ts[j]; li[tid * TOPK + j] = ti[j]; }
    __syncthreads();

    // log-tree merge of sorted descending 8-lists
    for (int n = 128; n; n >>= 1) {
        if (tid < n) {
            const int pa = tid * TOPK, pb = (tid + n) * TOPK;
            float mv[TOPK]; int mi[TOPK];
            int i = 0, j = 0;
#pragma unroll
            for (int k = 0; k < TOPK; ++k) {
                float va = ls[pa + i], vb = ls[pb + j];
                bool ta = va >= vb;
                mv[k] = ta ? va : vb;
                mi[k] = ta ? li[pa + i] : li[pb + j];
                i += ta ? 1 : 0; j += ta ? 0 : 1;
            }
#pragma unroll
            for (int k = 0; k < TOPK; ++k) { ls[pa + k] = mv[k]; li[pa + k] = mi[k]; }
        }
        __syncthreads();
    }

    float sc[TOPK]; int ix[TOPK];
#pragma unroll
    for (int k = 0; k < TOPK; ++k) { sc[k] = ls[k]; ix[k] = li[k]; }
    __syncthreads();                 // done reading merge result; ls reused below

    if (tid < TOPK) out[BB * DD + b * TOPK + tid] = sc[tid];   // scores (sorted desc)

    // softmax over 8 scores (sorted: max = sc[0])
    float w[TOPK], sum = 0.f;
#pragma unroll
    for (int k = 0; k < TOPK; ++k) { w[k] = expf(sc[k] - sc[0]); sum += w[k]; }
    const float inv = 1.0f / sum;
#pragma unroll
    for (int k = 0; k < TOPK; ++k) w[k] *= inv;

    // fused[d] = sum_k w_k * doc[ix_k][d] ; each thread owns d=tid and d=tid+256
    float f0 = 0.f, f1 = 0.f;
#pragma unroll
    for (int k = 0; k < TOPK; ++k) {
        const float* dr = doc + (size_t)ix[k] * DD;
        f0 = fmaf(w[k], dr[tid], f0);
        f1 = fmaf(w[k], dr[tid + 256], f1);
    }
    // gate = sigmoid(dot(fused, Wg) + bg)
    ls[tid] = f0 * Wg[tid] + f1 * Wg[tid + 256];
    __syncthreads();
    for (int n = 128; n; n >>= 1) {
        if (tid < n) ls[tid] += ls[tid + n];
        __syncthreads();
    }
    const float gate = 1.0f / (1.0f + expf(-(ls[0] + bg[0])));
    out[b * DD + tid]       = gate * f0;
    out[b * DD + tid + 256] = gate * f1;
}

// ------------------------------------------------------------------
extern "C" void kernel_launch(void* const* d_in, const int* in_sizes, int n_in,
                              void* d_out, int out_size, void* d_ws, size_t ws_size,
                              hipStream_t stream) {
    (void)in_sizes; (void)n_in; (void)out_size; (void)ws_size;
    const float* query = (const float*)d_in[0];   // [16,512,512]
    const float* doc   = (const float*)d_in[1];   // [500000,512]
    const float* W1    = (const float*)d_in[2];   // [512,512]
    const float* b1    = (const float*)d_in[3];
    const float* W2    = (const float*)d_in[4];
    const float* b2    = (const float*)d_in[5];
    const float* Wg    = (const float*)d_in[6];   // [1,512]
    const float* bg    = (const float*)d_in[7];   // [1]
    float* out = (float*)d_out;                   // evidence[16*512] ++ scores[16*8]

    float* ws     = (float*)d_ws;
    float* part   = ws;                 //  65536 floats
    float* hbar   = ws + 65536;         //   8192
    float* qvec   = ws + 73728;         //   8192
    float* qn     = ws + 81920;         //   8192
    float* cand_s = ws + 90112;         // 524288
    int*   cand_i = (int*)(ws + 614400);// 524288  (total ~4.56 MB)

    k1_mlp1      <<<dim3(4, 128), 128, 0, stream>>>(query, W1, b1, part);
    k1b_reduce   <<<32, 256, 0, stream>>>(part, hbar);
    k2_mlp2      <<<32, 32, 0, stream>>>(hbar, W2, b2, qvec);
    k2b_norm     <<<16, 256, 0, stream>>>(qvec, qn);
    k3_sims      <<<K3_BLOCKS, 256, 0, stream>>>(qn, doc, cand_s, cand_i);
    k4_merge_fuse<<<16, 256, 0, stream>>>(cand_s, cand_i, doc, Wg, bg, out);
}